// DoubleNet_53712861003989
// MI455X (gfx1250) — compile-verified
//
#include <hip/hip_runtime.h>
#include <hip/hip_bf16.h>

#define N_NODES   50000
#define N_FEAT    128
#define HIDDEN    128
#define N_CLASSES 40
#define NC_PAD    48          // N_CLASSES padded to multiple of 16 for WMMA
#define N_EDGES   1600000
#define N_POS     100000
#define N_NEG     100000

typedef __attribute__((ext_vector_type(2))) float v2f;
typedef __attribute__((ext_vector_type(8))) float v8f;

// ---------------------------------------------------------------- utilities

__global__ void zero_f32(float* __restrict__ p, long long n) {
  long long i = (long long)blockIdx.x * blockDim.x + threadIdx.x;
  long long stride = (long long)gridDim.x * blockDim.x;
  for (; i < n; i += stride) p[i] = 0.0f;
}

// Zero-pad W [K x N] into Wp [K x Np]
__global__ void pad_weights(const float* __restrict__ W, float* __restrict__ Wp,
                            int K, int N, int Np) {
  int i = blockIdx.x * blockDim.x + threadIdx.x;
  const int total = K * Np;
  const int stride = gridDim.x * blockDim.x;
  for (; i < total; i += stride) {
    const int k = i / Np, n = i - k * Np;
    Wp[i] = (n < N) ? W[k * N + n] : 0.0f;
  }
}

// ------------------------------------------------- WMMA fp32 GEMM  Y = X @ W
// Requirements: M % 16 == 0, N % 16 == 0, K == 128.  One wave per 16x16 tile,
// fully unrolled K loop of 32 V_WMMA_F32_16X16X4_F32.
// A 16x4 layout (ISA 7.12.2): lane m=lane&15, VGPR0/1 = K = 2*half / 2*half+1.
// C/D: VGPR r -> row r + 8*half, col = lane&15.
__global__ void gemm_wmma_f32(const float* __restrict__ X,
                              const float* __restrict__ W,
                              float* __restrict__ Y,
                              int M, int N) {
  constexpr int K = 128;
  const int lane = threadIdx.x;      // blockDim.x == 32
  const int half = lane >> 4;
  const int l16  = lane & 15;
  const long long row0 = (long long)blockIdx.x * 16;
  const int col0 = blockIdx.y * 16;

  const float* Arow = X + (row0 + l16) * K + 2 * half;      // 8B aligned
  const float* Bcol = W + (long long)(2 * half) * N + col0 + l16;

  v8f acc = {};
#pragma unroll
  for (int k = 0; k < K; k += 4) {
    const v2f a = *(const v2f*)(Arow + k);
    v2f b;
    b.x = Bcol[(long long)k * N];
    b.y = Bcol[(long long)k * N + N];
    acc = __builtin_amdgcn_wmma_f32_16x16x4_f32(
        false, a, false, b, (short)0, acc, false, false);
  }

  float* Yt = Y + (row0 + 8 * half) * N + col0 + l16;
#pragma unroll
  for (int r = 0; r < 8; ++r) Yt[(long long)r * N] = acc[r];
}

// ------------------------------------------------------------ degree / norm

__global__ void deg_accum(const int* __restrict__ cols,
                          const float* __restrict__ w,  // nullptr => weight 1
                          float* __restrict__ deg, int E) {
  int i = blockIdx.x * blockDim.x + threadIdx.x;
  const int stride = gridDim.x * blockDim.x;
  for (; i < E; i += stride) {
    const float wv = w ? w[i] : 1.0f;
    if (wv != 0.0f) atomicAdd(&deg[cols[i]], wv);
  }
}

__global__ void dinv_kernel(const float* __restrict__ deg,
                            float* __restrict__ dinv, int n) {
  int i = blockIdx.x * blockDim.x + threadIdx.x;
  const int stride = gridDim.x * blockDim.x;
  for (; i < n; i += stride) {
    const float d = deg[i] + 1.0f;   // + self loop
    dinv[i] = (d > 0.0f) ? rsqrtf(d) : 0.0f;
  }
}

// ------------------------------------------------ edge scatter (segment sum)
// acc[c][0..d) += dinv[r]*w*dinv[c] * XW[r][0..d).  One thread = one edge x 4
// features: float4 gather of XW (16B aligned for ld in {128,48}), 4 atomics.
__global__ void scatter_edges(const int* __restrict__ rows,
                              const int* __restrict__ cols,
                              const float* __restrict__ w,   // nullptr => 1
                              const float* __restrict__ dinv,
                              const float* __restrict__ XW, int ldx,
                              float* __restrict__ acc, int d,
                              int E) {
  const int d4 = d >> 2;
  long long i = (long long)blockIdx.x * blockDim.x + threadIdx.x;
  const long long stride = (long long)gridDim.x * blockDim.x;
  const long long total = (long long)E * d4;
  for (; i < total; i += stride) {
    const int e  = (int)(i / d4);
    const int f4 = (int)(i - (long long)e * d4);
    const float wv = w ? w[e] : 1.0f;
    if (wv == 0.0f) continue;
    const int r = rows[e];
    const int c = cols[e];
    const float nrm = dinv[r] * wv * dinv[c];
    const float4 v = *(const float4*)(XW + (long long)r * ldx + 4 * f4);
    float* dst = acc + (long long)c * d + 4 * f4;
    atomicAdd(dst + 0, nrm * v.x);
    atomicAdd(dst + 1, nrm * v.y);
    atomicAdd(dst + 2, nrm * v.z);
    atomicAdd(dst + 3, nrm * v.w);
  }
}

// ---------------------------- finalize: self-loop term + bias (+ReLU opt.)
__global__ void finalize_conv(const float* __restrict__ acc,
                              const float* __restrict__ XW, int ldx,
                              const float* __restrict__ dinv,
                              const float* __restrict__ bias,
                              float* __restrict__ out,
                              int n, int d, int relu) {
  long long i = (long long)blockIdx.x * blockDim.x + threadIdx.x;
  const long long stride = (long long)gridDim.x * blockDim.x;
  const long long total = (long long)n * d;
  for (; i < total; i += stride) {
    const int node = (int)(i / d);
    const int f = (int)(i - (long long)node * d);
    const float di = dinv[node];
    float v = acc[i] + di * di * XW[(long long)node * ldx + f] + bias[f];
    if (relu) v = fmaxf(v, 0.0f);
    out[i] = v;
  }
}

// ------------------------------- decoder: m[j] = (dot(z[tr],z[tc]) > 0)
// One wave per candidate; each lane loads exactly one float4 of each z row.
__global__ void decoder_kernel(const float* __restrict__ z,
                               const int* __restrict__ pos,
                               const int* __restrict__ neg,
                               float* __restrict__ m,
                               int* __restrict__ count) {
  const int wpb = blockDim.x >> 5;
  const int j = blockIdx.x * wpb + (threadIdx.x >> 5);
  const int lane = threadIdx.x & 31;
  if (j >= N_POS + N_NEG) return;
  int s, t;
  if (j < N_POS) { s = pos[j];          t = pos[N_POS + j]; }
  else           { s = neg[j - N_POS];  t = neg[N_NEG + (j - N_POS)]; }
  const float4 zs = ((const float4*)(z + (long long)s * HIDDEN))[lane];
  const float4 zt = ((const float4*)(z + (long long)t * HIDDEN))[lane];
  float a = zs.x * zt.x + zs.y * zt.y + zs.z * zt.z + zs.w * zt.w;
  for (int o = 16; o > 0; o >>= 1) a += __shfl_xor(a, o, 32);
  if (lane == 0) {
    const float mv = (a > 0.0f) ? 1.0f : 0.0f;
    m[j] = mv;
    if (mv != 0.0f) atomicAdd(count, 1);
  }
}

// --------------------------------------------- per-row log_softmax (d = 40)
__global__ void log_softmax_rows(const float* __restrict__ logits,
                                 float* __restrict__ out, int n, int d) {
  const int wpb = blockDim.x >> 5;
  const int row = blockIdx.x * wpb + (threadIdx.x >> 5);
  const int lane = threadIdx.x & 31;
  if (row >= n) return;
  const float* p = logits + (long long)row * d;
  float mx = -__builtin_inff();
  for (int f = lane; f < d; f += 32) mx = fmaxf(mx, p[f]);
  for (int o = 16; o > 0; o >>= 1) mx = fmaxf(mx, __shfl_xor(mx, o, 32));
  float s = 0.0f;
  for (int f = lane; f < d; f += 32) s += __expf(p[f] - mx);
  for (int o = 16; o > 0; o >>= 1) s += __shfl_xor(s, o, 32);
  const float lse = mx + __logf(s);
  for (int f = lane; f < d; f += 32)
    out[(long long)row * d + f] = p[f] - lse;
}

__global__ void write_count(const int* __restrict__ count,
                            float* __restrict__ dst) {
  if (blockIdx.x == 0 && threadIdx.x == 0) dst[0] = (float)count[0];
}

// ------------------------------------------------------------------- host

static inline int nblk(long long n, int bs, int cap) {
  long long b = (n + bs - 1) / bs;
  return (int)(b < (long long)cap ? (b > 0 ? b : 1) : cap);
}

extern "C" void kernel_launch(void* const* d_in, const int* in_sizes, int n_in,
                              void* d_out, int out_size, void* d_ws, size_t ws_size,
                              hipStream_t stream) {
  // inputs (setup_inputs order)
  const float* x   = (const float*)d_in[0];
  const float* W1  = (const float*)d_in[1];
  const float* b1  = (const float*)d_in[2];
  const float* W2  = (const float*)d_in[3];
  const float* b2  = (const float*)d_in[4];
  const float* W1s = (const float*)d_in[5];
  const float* b1s = (const float*)d_in[6];
  const float* W2s = (const float*)d_in[7];
  const float* b2s = (const float*)d_in[8];
  const int* ei  = (const int*)d_in[9];    // [2, N_EDGES]
  const int* pei = (const int*)d_in[10];   // [2, N_POS]
  const int* nei = (const int*)d_in[11];   // [2, N_NEG]
  const int* erow = ei;
  const int* ecol = ei + N_EDGES;

  // workspace layout (floats)
  float* ws = (float*)d_ws;
  size_t o = 0;
  float* B_xw  = ws + o; o += (size_t)N_NODES * HIDDEN;   // GEMM output buffer
  float* B_acc = ws + o; o += (size_t)N_NODES * HIDDEN;   // aggregation accumulator
  float* B_h   = ws + o; o += (size_t)N_NODES * HIDDEN;   // h, then h2
  float* deg1  = ws + o; o += N_NODES;
  float* dinv1 = ws + o; o += N_NODES;
  float* deg2  = ws + o; o += N_NODES;
  float* dinv2 = ws + o; o += N_NODES;
  float* mbuf  = ws + o; o += (size_t)(N_POS + N_NEG);
  float* W2sp  = ws + o; o += (size_t)HIDDEN * NC_PAD;    // padded W2s
  int*   count = (int*)(ws + o); o += 1;

  // output layout: log_softmax[50000*40] | z[50000*128] | n_selected
  float* out_ls = (float*)d_out;
  float* zreg   = out_ls + (size_t)N_NODES * N_CLASSES;
  float* nsel   = zreg + (size_t)N_NODES * HIDDEN;

  const int TB = 256;
  const long long ND  = (long long)N_NODES * HIDDEN;     // 6.4M
  const long long NC  = (long long)N_NODES * N_CLASSES;  // 2.0M

  // ---------------- graph 1: degree / dinv ----------------
  zero_f32<<<nblk(N_NODES, TB, 4096), TB, 0, stream>>>(deg1, N_NODES);
  deg_accum<<<nblk(N_EDGES, TB, 4096), TB, 0, stream>>>(ecol, nullptr, deg1, N_EDGES);
  dinv_kernel<<<nblk(N_NODES, TB, 4096), TB, 0, stream>>>(deg1, dinv1, N_NODES);

  // ---------------- conv1: h = relu(gcn(x, W1, b1)) ----------------
  gemm_wmma_f32<<<dim3(N_NODES / 16, HIDDEN / 16), 32, 0, stream>>>(
      x, W1, B_xw, N_NODES, HIDDEN);
  zero_f32<<<nblk(ND, TB, 8192), TB, 0, stream>>>(B_acc, ND);
  scatter_edges<<<8192, TB, 0, stream>>>(erow, ecol, nullptr, dinv1,
                                         B_xw, HIDDEN, B_acc, HIDDEN, N_EDGES);
  finalize_conv<<<nblk(ND, TB, 8192), TB, 0, stream>>>(
      B_acc, B_xw, HIDDEN, dinv1, b1, B_h, N_NODES, HIDDEN, 1);

  // ---------------- conv2: z = gcn(h, W2, b2)  (into d_out z region) -------
  gemm_wmma_f32<<<dim3(N_NODES / 16, HIDDEN / 16), 32, 0, stream>>>(
      B_h, W2, B_xw, N_NODES, HIDDEN);
  zero_f32<<<nblk(ND, TB, 8192), TB, 0, stream>>>(zreg, ND);
  scatter_edges<<<8192, TB, 0, stream>>>(erow, ecol, nullptr, dinv1,
                                         B_xw, HIDDEN, zreg, HIDDEN, N_EDGES);
  finalize_conv<<<nblk(ND, TB, 8192), TB, 0, stream>>>(
      zreg, B_xw, HIDDEN, dinv1, b2, zreg, N_NODES, HIDDEN, 0);

  // ---------------- decoder: m = (z[tr].z[tc] > 0) ----------------
  zero_f32<<<1, 32, 0, stream>>>((float*)count, 1);
  decoder_kernel<<<(N_POS + N_NEG + 7) / 8, 256, 0, stream>>>(zreg, pei, nei,
                                                              mbuf, count);

  // ---------------- graph 2: degree / dinv ----------------
  zero_f32<<<nblk(N_NODES, TB, 4096), TB, 0, stream>>>(deg2, N_NODES);
  deg_accum<<<nblk(N_EDGES, TB, 4096), TB, 0, stream>>>(ecol, nullptr, deg2, N_EDGES);
  deg_accum<<<nblk(N_POS, TB, 4096), TB, 0, stream>>>(pei + N_POS, mbuf,         deg2, N_POS);
  deg_accum<<<nblk(N_NEG, TB, 4096), TB, 0, stream>>>(nei + N_NEG, mbuf + N_POS, deg2, N_NEG);
  deg_accum<<<nblk(N_POS, TB, 4096), TB, 0, stream>>>(pei,         mbuf,         deg2, N_POS);
  deg_accum<<<nblk(N_NEG, TB, 4096), TB, 0, stream>>>(nei,         mbuf + N_POS, deg2, N_NEG);
  dinv_kernel<<<nblk(N_NODES, TB, 4096), TB, 0, stream>>>(deg2, dinv2, N_NODES);

  // ---------------- conv3: h2 = relu(gcn(x, W1s, b1s, rewired)) ------------
  gemm_wmma_f32<<<dim3(N_NODES / 16, HIDDEN / 16), 32, 0, stream>>>(
      x, W1s, B_xw, N_NODES, HIDDEN);
  zero_f32<<<nblk(ND, TB, 8192), TB, 0, stream>>>(B_acc, ND);
  scatter_edges<<<8192, TB, 0, stream>>>(erow, ecol, nullptr, dinv2,
                                         B_xw, HIDDEN, B_acc, HIDDEN, N_EDGES);
  scatter_edges<<<4096, TB, 0, stream>>>(pei, pei + N_POS, mbuf,         dinv2, B_xw, HIDDEN, B_acc, HIDDEN, N_POS);
  scatter_edges<<<4096, TB, 0, stream>>>(nei, nei + N_NEG, mbuf + N_POS, dinv2, B_xw, HIDDEN, B_acc, HIDDEN, N_NEG);
  scatter_edges<<<4096, TB, 0, stream>>>(pei + N_POS, pei, mbuf,         dinv2, B_xw, HIDDEN, B_acc, HIDDEN, N_POS);
  scatter_edges<<<4096, TB, 0, stream>>>(nei + N_NEG, nei, mbuf + N_POS, dinv2, B_xw, HIDDEN, B_acc, HIDDEN, N_NEG);
  finalize_conv<<<nblk(ND, TB, 8192), TB, 0, stream>>>(
      B_acc, B_xw, HIDDEN, dinv2, b1s, B_h, N_NODES, HIDDEN, 1);

  // ---------------- conv4: out = gcn(h2, W2s, b2s, rewired) ----------------
  pad_weights<<<nblk(HIDDEN * NC_PAD, TB, 64), TB, 0, stream>>>(
      W2s, W2sp, HIDDEN, N_CLASSES, NC_PAD);
  gemm_wmma_f32<<<dim3(N_NODES / 16, NC_PAD / 16), 32, 0, stream>>>(
      B_h, W2sp, B_xw, N_NODES, NC_PAD);              // XW with ld = 48
  zero_f32<<<nblk(NC, TB, 8192), TB, 0, stream>>>(B_acc, NC);
  scatter_edges<<<8192, TB, 0, stream>>>(erow, ecol, nullptr, dinv2,
                                         B_xw, NC_PAD, B_acc, N_CLASSES, N_EDGES);
  scatter_edges<<<4096, TB, 0, stream>>>(pei, pei + N_POS, mbuf,         dinv2, B_xw, NC_PAD, B_acc, N_CLASSES, N_POS);
  scatter_edges<<<4096, TB, 0, stream>>>(nei, nei + N_NEG, mbuf + N_POS, dinv2, B_xw, NC_PAD, B_acc, N_CLASSES, N_NEG);
  scatter_edges<<<4096, TB, 0, stream>>>(pei + N_POS, pei, mbuf,         dinv2, B_xw, NC_PAD, B_acc, N_CLASSES, N_POS);
  scatter_edges<<<4096, TB, 0, stream>>>(nei + N_NEG, nei, mbuf + N_POS, dinv2, B_xw, NC_PAD, B_acc, N_CLASSES, N_NEG);
  finalize_conv<<<nblk(NC, TB, 8192), TB, 0, stream>>>(
      B_acc, B_xw, NC_PAD, dinv2, b2s, B_acc, N_NODES, N_CLASSES, 0);

  // ---------------- outputs ----------------
  log_softmax_rows<<<(N_NODES + 7) / 8, 256, 0, stream>>>(B_acc, out_ls,
                                                          N_NODES, N_CLASSES);
  write_count<<<1, 32, 0, stream>>>(count, nsel);
}